// DCN_model_6665789243911
// MI455X (gfx1250) — compile-verified
//
#include <hip/hip_runtime.h>
#include <math.h>

// ---------------------------------------------------------------------------
// DCNv2 x3 for MI455X (gfx1250): fused bilinear-im2col + FP32 WMMA GEMM.
// All matrix math uses V_WMMA_F32_16X16X4_F32 (full fp32 matrix path).
// Templated on channel count so all per-chunk index math constant-folds
// (no runtime scalar division in the hot loop), with the per-tap bilinear
// setup hoisted out of the channel-chunk loop.
// ---------------------------------------------------------------------------

typedef __attribute__((ext_vector_type(2))) float v2f;
typedef __attribute__((ext_vector_type(8))) float v8f;

#define BATCH 4
#define HH    96
#define WW_   96
#define HWSZ  (HH * WW_)          // 9216
#define PTOT  (BATCH * HWSZ)      // 36864

__device__ __forceinline__ v8f wmma_f32(v2f a, v2f b, v8f c) {
#if defined(__gfx1250__)
  // 8 args: (neg_a, A, neg_b, B, c_mod, C, reuse_a, reuse_b)
  return __builtin_amdgcn_wmma_f32_16x16x4_f32(false, a, false, b, (short)0, c,
                                               false, false);
#else
  (void)a; (void)b;
  return c;  // host pass / non-gfx1250: never executed
#endif
}

// ---------------------------------------------------------------------------
// Offset conv: om[b, 0..26, h, w] = conv3x3(img, ow) + ob
// GEMM view: [32(pad 27) x 9C] x [9C x P].  Block: 256 thr = 8 waves.
// Wave w: O-tile = w&1, P-subtile = w>>1 (0..3).  P-tile = 64, K-chunk = 64.
// ---------------------------------------------------------------------------
template <int C>
__global__ void off_conv(const float* __restrict__ img,
                         const float* __restrict__ ow,
                         const float* __restrict__ ob,
                         float* __restrict__ om) {
  __shared__ float smp[64][64];   // [k-row][pixel]
  __shared__ float wts[32][66];   // padded stride -> conflict-free A reads

  constexpr int NTHR = 256;
  constexpr int NCC  = C / 64;    // channel chunks per tap

  const int tid  = threadIdx.x;
  const int lane = tid & 31;
  const int wav  = tid >> 5;            // 0..7
  const int ot   = wav & 1;             // O tile 0/1
  const int pt   = wav >> 1;            // P subtile 0..3
  const int m16  = lane & 15;
  const int kh   = lane >> 4;           // lane half

  const int pblk = blockIdx.x * 64;     // HW % 64 == 0 -> whole WG same batch
  const int b    = pblk / HWSZ;
  const int qb   = pblk - b * HWSZ;

  const int p_l  = tid & 63;            // this thread's pixel column
  const int q    = qb + p_l;
  const int h    = q / WW_;
  const int wcol = q - h * WW_;

  const float* imgB = img + (size_t)b * C * HWSZ;

  v8f acc = {};

  for (int k = 0; k < 9; ++k) {
    const int dy = k / 3 - 1;
    const int dx = k - 3 * (k / 3) - 1;

    // ---- per-tap shifted-window setup (once per tap) ----
    const int  hy    = h + dy;
    const int  wx    = wcol + dx;
    const bool valid = ((unsigned)hy < (unsigned)HH) &&
                       ((unsigned)wx < (unsigned)WW_);
    const int  lin   = valid ? (hy * WW_ + wx) : 0;

    for (int cc = 0; cc < NCC; ++cc) {
      const int cbase = cc * 64;

      __syncthreads();
      // ---- stage shifted input (zero pad) ----
      {
        const float* ib = imgB + (size_t)cbase * HWSZ;
#pragma unroll
        for (int c_l = tid >> 6; c_l < 64; c_l += (NTHR >> 6)) {
          smp[c_l][p_l] = valid ? ib[(size_t)c_l * HWSZ + lin] : 0.0f;
        }
      }
      // ---- stage weights (pad rows 27..31 with zero) ----
#pragma unroll
      for (int i = tid; i < 32 * 64; i += NTHR) {
        const int o_l = i >> 6;
        const int c_l = i & 63;
        float v = 0.0f;
        if (o_l < 27) v = ow[((size_t)o_l * C + cbase + c_l) * 9 + k];
        wts[o_l][c_l] = v;
      }
      __syncthreads();

      // ---- 16 WMMA steps over the K-chunk ----
#pragma unroll
      for (int kk = 0; kk < 64; kk += 4) {
        const int kr = kk + 2 * kh;
        v2f a, bb;
        a.x  = wts[ot * 16 + m16][kr];
        a.y  = wts[ot * 16 + m16][kr + 1];
        bb.x = smp[kr][pt * 16 + m16];
        bb.y = smp[kr + 1][pt * 16 + m16];
        acc  = wmma_f32(a, bb, acc);
      }
    }
  }

  // ---- epilogue: rows M=v (lane half 0) / v+8 (half 1) ----
#pragma unroll
  for (int v = 0; v < 8; ++v) {
    const int o = ot * 16 + v + 8 * kh;
    if (o < 27) {
      om[((size_t)b * 27 + o) * HWSZ + qb + pt * 16 + m16] = acc[v] + ob[o];
    }
  }
}

// ---------------------------------------------------------------------------
// Main DCN: out[b,o,h,w] = sum_{k,c} w[o,c,k] * bilinear(img, p+off_k) * mask_k
// GEMM view: [O x 9C] x [9C x P], column matrix generated on the fly in LDS.
// NW waves; wave w owns O-tile w and two P-subtiles (P-tile = 32). K-chunk 64.
// Bilinear coords/corner weights computed once per tap, reused for all
// channel chunks of that tap.
// ---------------------------------------------------------------------------
template <int C, int NW, int O>
__global__ void dcn_main(const float* __restrict__ img,
                         const float* __restrict__ om,
                         const float* __restrict__ wmat,
                         const float* __restrict__ bias,
                         float* __restrict__ out) {
  __shared__ float smp[64][32];        // sampled column tile [k-row][pixel]
  __shared__ float wts[NW * 16][66];   // weight tile, padded stride

  constexpr int NTHR = NW * 32;
  constexpr int OWG  = NW * 16;        // O rows per workgroup
  constexpr int NCC  = C / 64;         // channel chunks per tap

  const int tid  = threadIdx.x;
  const int lane = tid & 31;
  const int ot   = tid >> 5;           // wave id == O tile
  const int m16  = lane & 15;
  const int kh   = lane >> 4;

  const int oblk = blockIdx.y * OWG;

  const int pblk = blockIdx.x * 32;    // HW % 32 == 0 -> same batch in WG
  const int b    = pblk / HWSZ;
  const int qb   = pblk - b * HWSZ;

  const int p_l  = tid & 31;
  const int q    = qb + p_l;
  const int h    = q / WW_;
  const int wcol = q - h * WW_;

  v8f acc0 = {};
  v8f acc1 = {};

  const float* omB  = om + (size_t)b * 27 * HWSZ;
  const float* imgB = img + (size_t)b * C * HWSZ;

  for (int k = 0; k < 9; ++k) {
    const int dy = k / 3 - 1;
    const int dx = k - 3 * (k / 3) - 1;

    // ---- per-tap bilinear setup (once per tap, reused across chunks) ----
    const float oy = omB[(size_t)(2 * k) * HWSZ + q];
    const float ox = omB[(size_t)(2 * k + 1) * HWSZ + q];
    float mm       = omB[(size_t)(18 + k) * HWSZ + q];
    mm = 1.0f / (1.0f + __expf(-mm));   // sigmoid mask

    const float py  = (float)(h + dy) + oy;
    const float px  = (float)(wcol + dx) + ox;
    const float y0f = floorf(py);
    const float x0f = floorf(px);
    const int   y0  = (int)y0f;
    const int   x0  = (int)x0f;
    const float fy  = py - y0f;
    const float fx  = px - x0f;

    const bool vy0 = (unsigned)y0 < (unsigned)HH;
    const bool vy1 = (unsigned)(y0 + 1) < (unsigned)HH;
    const bool vx0 = (unsigned)x0 < (unsigned)WW_;
    const bool vx1 = (unsigned)(x0 + 1) < (unsigned)WW_;

    // fold mask + validity into the 4 corner weights
    const float w00 = (vy0 && vx0) ? (1.0f - fy) * (1.0f - fx) * mm : 0.0f;
    const float w01 = (vy0 && vx1) ? (1.0f - fy) * fx * mm : 0.0f;
    const float w10 = (vy1 && vx0) ? fy * (1.0f - fx) * mm : 0.0f;
    const float w11 = (vy1 && vx1) ? fy * fx * mm : 0.0f;

    const int cy0 = min(max(y0, 0), HH - 1);
    const int cy1 = min(max(y0 + 1, 0), HH - 1);
    const int cx0 = min(max(x0, 0), WW_ - 1);
    const int cx1 = min(max(x0 + 1, 0), WW_ - 1);
    const int l00 = cy0 * WW_ + cx0;
    const int l01 = cy0 * WW_ + cx1;
    const int l10 = cy1 * WW_ + cx0;
    const int l11 = cy1 * WW_ + cx1;

    for (int cc = 0; cc < NCC; ++cc) {
      const int cbase = cc * 64;

      __syncthreads();
      // ---- bilinear sample 64 channels x 32 pixels into LDS ----
      {
        const float* ib = imgB + (size_t)cbase * HWSZ;
#pragma unroll
        for (int c_l = tid >> 5; c_l < 64; c_l += NW) {
          const float* pc = ib + (size_t)c_l * HWSZ;
          smp[c_l][p_l] = w00 * pc[l00] + w01 * pc[l01] +
                          w10 * pc[l10] + w11 * pc[l11];
        }
      }
      // ---- stage weight tile ----
#pragma unroll
      for (int i = tid; i < (OWG << 6); i += NTHR) {
        const int o_l = i >> 6;
        const int c_l = i & 63;
        wts[o_l][c_l] = wmat[((size_t)(oblk + o_l) * C + cbase + c_l) * 9 + k];
      }
      __syncthreads();

      // ---- 16 WMMA steps x 2 P-subtiles (A reused) ----
#pragma unroll
      for (int kk = 0; kk < 64; kk += 4) {
        const int kr = kk + 2 * kh;
        v2f a, b0, b1;
        a.x  = wts[ot * 16 + m16][kr];
        a.y  = wts[ot * 16 + m16][kr + 1];
        b0.x = smp[kr][m16];
        b0.y = smp[kr + 1][m16];
        b1.x = smp[kr][16 + m16];
        b1.y = smp[kr + 1][16 + m16];
        acc0 = wmma_f32(a, b0, acc0);
        acc1 = wmma_f32(a, b1, acc1);
      }
    }
  }

  // ---- epilogue: C/D layout rows M=v / v+8 per lane half; add bias ----
#pragma unroll
  for (int v = 0; v < 8; ++v) {
    const int o  = oblk + ot * 16 + v + 8 * kh;
    const float bs = bias[o];
    float* op = out + ((size_t)b * O + o) * HWSZ + qb;
    op[m16]      = acc0[v] + bs;
    op[16 + m16] = acc1[v] + bs;
  }
}

// ---------------------------------------------------------------------------
// Driver: 3 x (offset-conv -> fused DCN GEMM), stream-ordered, capture-safe.
// Workspace: om(27P) | y1(64P) | y2(512P) floats  ~= 89 MB.
// ---------------------------------------------------------------------------
extern "C" void kernel_launch(void* const* d_in, const int* in_sizes, int n_in,
                              void* d_out, int out_size, void* d_ws,
                              size_t ws_size, hipStream_t stream) {
  (void)in_sizes; (void)n_in; (void)out_size; (void)ws_size;

  const float* x   = (const float*)d_in[0];
  const float* ow1 = (const float*)d_in[1];
  const float* ob1 = (const float*)d_in[2];
  const float* w1  = (const float*)d_in[3];
  const float* b1  = (const float*)d_in[4];
  const float* ow2 = (const float*)d_in[5];
  const float* ob2 = (const float*)d_in[6];
  const float* w2  = (const float*)d_in[7];
  const float* b2  = (const float*)d_in[8];
  const float* ow3 = (const float*)d_in[9];
  const float* ob3 = (const float*)d_in[10];
  const float* w3  = (const float*)d_in[11];
  const float* b3  = (const float*)d_in[12];

  float* om = (float*)d_ws;
  float* y1 = om + (size_t)27 * PTOT;
  float* y2 = y1 + (size_t)64 * PTOT;
  float* y3 = (float*)d_out;

  const dim3 gOff(PTOT / 64, 1, 1);

  // Layer 1: C=64 -> O=64
  off_conv<64><<<gOff, 256, 0, stream>>>(x, ow1, ob1, om);
  dcn_main<64, 4, 64><<<dim3(PTOT / 32, 1, 1), 128, 0, stream>>>(
      x, om, w1, b1, y1);

  // Layer 2: C=64 -> O=512
  off_conv<64><<<gOff, 256, 0, stream>>>(y1, ow2, ob2, om);
  dcn_main<64, 8, 512><<<dim3(PTOT / 32, 4, 1), 256, 0, stream>>>(
      y1, om, w2, b2, y2);

  // Layer 3: C=512 -> O=256
  off_conv<512><<<gOff, 256, 0, stream>>>(y2, ow3, ob3, om);
  dcn_main<512, 8, 256><<<dim3(PTOT / 32, 2, 1), 256, 0, stream>>>(
      y2, om, w3, b3, y3);
}